// MultiheadAttention_9594956939915
// MI455X (gfx1250) — compile-verified
//
#include <hip/hip_runtime.h>
#include <hip/hip_bf16.h>

// Problem constants (match reference)
#define B_  256
#define N_  2048
#define D_  512
#define H_  8
#define DH_ 64
#define NK  2047   // N-1 keys/values per row

typedef float v2f __attribute__((ext_vector_type(2)));
typedef float v8f __attribute__((ext_vector_type(8)));

static __device__ __forceinline__ v8f wmma4_f32(v2f a, v2f b, v8f c) {
  // D = A(16x4 f32) x B(4x16 f32) + C(16x16 f32)
  return __builtin_amdgcn_wmma_f32_16x16x4_f32(false, a, false, b, (short)0, c,
                                               false, false);
}

// ---------------------------------------------------------------------------
// Kernel 1: q[h,b,k] = (1/sqrt(DH)) * sum_d ego[b,d] * W_q[h,d,k]
// One wave per (head, 16-row b-tile). fp32 WMMA 16x16x4, K-loop over D.
// ---------------------------------------------------------------------------
__global__ __launch_bounds__(32) void qproj_wmma(const float* __restrict__ ego,
                                                 const float* __restrict__ wq,
                                                 float* __restrict__ qout) {
  const int lane = threadIdx.x;        // 0..31
  const int lc   = lane & 15;
  const int half = lane >> 4;
  const int b0   = blockIdx.x * 16;
  const int h    = blockIdx.y;

  const float* wh = wq + (size_t)h * D_ * DH_;   // W_q[h] : (D, DH)
  v8f acc0 = {}, acc1 = {}, acc2 = {}, acc3 = {};

  for (int d0 = 0; d0 < D_; d0 += 4) {
    // A fragment: ego tile (16 rows b x 4 cols d)
    const float* er = ego + (size_t)(b0 + lc) * D_ + d0 + 2 * half;
    v2f a; a.x = er[0]; a.y = er[1];
    // B fragments: W_q chunk (4 d x 64 k) -> four 4x16 tiles
    const float* w0 = wh + (size_t)(d0 + 2 * half) * DH_ + lc;
    const float* w1 = w0 + DH_;
    v2f bf;
    bf.x = w0[0];  bf.y = w1[0];  acc0 = wmma4_f32(a, bf, acc0);
    bf.x = w0[16]; bf.y = w1[16]; acc1 = wmma4_f32(a, bf, acc1);
    bf.x = w0[32]; bf.y = w1[32]; acc2 = wmma4_f32(a, bf, acc2);
    bf.x = w0[48]; bf.y = w1[48]; acc3 = wmma4_f32(a, bf, acc3);
  }

  const float scale = 0.125f;  // 1/sqrt(64), folded into q
#pragma unroll
  for (int j = 0; j < 8; ++j) {
    // C/D layout: VGPR j holds row M=j (lanes 0-15) and M=j+8 (lanes 16-31)
    float* qr = qout + ((size_t)h * B_ + (b0 + j + 8 * half)) * DH_;
    qr[lc]      = acc0[j] * scale;
    qr[16 + lc] = acc1[j] * scale;
    qr[32 + lc] = acc2[j] * scale;
    qr[48 + lc] = acc3[j] * scale;
  }
}

// ---------------------------------------------------------------------------
// Kernel 2: fused masked-softmax attention, one block per batch row b.
// 8 waves = 8 heads. Streams keys (1.05 GB total) once, value (134 MB) once.
// ---------------------------------------------------------------------------
__global__ __launch_bounds__(256) void attn_fused(
    const float* __restrict__ q,      // (H,B,DH), pre-scaled by 1/sqrt(DH)
    const unsigned char* __restrict__ mask,  // (B,N) bool
    const float* __restrict__ keys,   // (H,B,NK,DH)
    const float* __restrict__ value,  // (B,NK,DH)
    float* __restrict__ out)          // (B, D) heads concatenated
{
  __shared__ float s_p[H_ * N_];          // scores -> probabilities (64 KB)
  __shared__ float s_part[H_ * H_ * DH_]; // per-wave PV partials (16 KB)
  __shared__ float s_rscale[H_];

  const int b    = blockIdx.x;
  const int tid  = threadIdx.x;
  const int wave = tid >> 5;   // 0..7
  const int lane = tid & 31;
  const int h    = wave;       // phase A/B1: wave == head

  // q[h][b][*] into registers (uniform per wave -> scalarizable load)
  const float4* qh = (const float4*)(q + ((size_t)h * B_ + b) * DH_);
  float4 qr[16];
#pragma unroll
  for (int i = 0; i < 16; ++i) qr[i] = qh[i];

  const float* kbase = keys + ((size_t)h * B_ + b) * (size_t)NK * DH_;
  const unsigned char* mrow = mask + (size_t)b * N_ + 1;

  // ---- Phase A: scores[h][n] = q.k, masked fill with -inf -----------------
  for (int it = 0; it < 64; ++it) {
    const int n = it * 32 + lane;     // consecutive n across the wave
    if (n < NK) {
      const float4* kp = (const float4*)(kbase + (size_t)n * DH_);
      if (n + 32 < NK)  // keep the HBM stream ahead (global_prefetch_b8)
        __builtin_prefetch(kbase + (size_t)(n + 32) * DH_, 0, 1);
      float acc = 0.f;
#pragma unroll
      for (int i = 0; i < 16; ++i) {
        const float4 kv = kp[i];
        acc = fmaf(qr[i].x, kv.x, acc);
        acc = fmaf(qr[i].y, kv.y, acc);
        acc = fmaf(qr[i].z, kv.z, acc);
        acc = fmaf(qr[i].w, kv.w, acc);
      }
      s_p[h * N_ + n] = mrow[n] ? acc : -__builtin_inff();
    } else {
      s_p[h * N_ + n] = 0.0f;   // n == 2047 pad column (zero prob for WMMA)
    }
  }
  __syncthreads();

  // ---- Phase B1: per-head softmax (wave32 shuffle reductions) -------------
  float m = -__builtin_inff();
  for (int n = lane; n < NK; n += 32) m = fmaxf(m, s_p[h * N_ + n]);
#pragma unroll
  for (int o = 16; o > 0; o >>= 1) m = fmaxf(m, __shfl_xor(m, o, 32));

  float ssum = 0.f;
  for (int n = lane; n < NK; n += 32) {
    const float p = __expf(s_p[h * N_ + n] - m);
    s_p[h * N_ + n] = p;
    ssum += p;
  }
#pragma unroll
  for (int o = 16; o > 0; o >>= 1) ssum += __shfl_xor(ssum, o, 32);
  if (lane == 0) s_rscale[h] = 1.0f / ssum;
  __syncthreads();

  // ---- Phase B2: out[h,:] = P(8x2047) x V(2047x64) via fp32 WMMA ----------
  // A = P tile (16 rows: heads 0-7, rows 8-15 zero) x 4 n; B = V chunk 4x16.
  // Each wave owns K-slice [wave*256, wave*256+256) (padded with zero prob).
  const int lc   = lane & 15;
  const int half = lane >> 4;
  const float* vbase = value + (size_t)b * NK * DH_;
  v8f acc0 = {}, acc1 = {}, acc2 = {}, acc3 = {};
  const int nw0 = wave * 256;

  for (int i = 0; i < 64; ++i) {
    const int n0 = nw0 + i * 4;
    const int r0 = n0 + 2 * half;              // this half-wave's K pair
    v2f a;
    a.x = (lc < H_) ? s_p[lc * N_ + r0]     : 0.f;
    a.y = (lc < H_) ? s_p[lc * N_ + r0 + 1] : 0.f;
    const int vr0 = (r0     < NK) ? r0     : NK - 1;  // clamp: p==0 there
    const int vr1 = (r0 + 1 < NK) ? r0 + 1 : NK - 1;
    const float* v0 = vbase + (size_t)vr0 * DH_;
    const float* v1 = vbase + (size_t)vr1 * DH_;
    v2f bf;
    bf.x = v0[lc];      bf.y = v1[lc];      acc0 = wmma4_f32(a, bf, acc0);
    bf.x = v0[16 + lc]; bf.y = v1[16 + lc]; acc1 = wmma4_f32(a, bf, acc1);
    bf.x = v0[32 + lc]; bf.y = v1[32 + lc]; acc2 = wmma4_f32(a, bf, acc2);
    bf.x = v0[48 + lc]; bf.y = v1[48 + lc]; acc3 = wmma4_f32(a, bf, acc3);
  }

  // Rows 0-7 (valid heads) live in VGPRs 0..7 of lanes 0-15.
  if (half == 0) {
    float* pp = s_part + wave * (H_ * DH_);
#pragma unroll
    for (int j = 0; j < 8; ++j) {
      pp[j * DH_ + lc]      = acc0[j];
      pp[j * DH_ + 16 + lc] = acc1[j];
      pp[j * DH_ + 32 + lc] = acc2[j];
      pp[j * DH_ + 48 + lc] = acc3[j];
    }
  }
  __syncthreads();

  // ---- Reduce 8 wave-partials, apply 1/sum, write (b, h*64+k) -------------
  for (int o = tid; o < D_; o += 256) {
    const int hh = o >> 6, k = o & 63;
    float s = 0.f;
#pragma unroll
    for (int w = 0; w < H_; ++w) s += s_part[w * (H_ * DH_) + hh * DH_ + k];
    out[(size_t)b * D_ + o] = s * s_rscale[hh];
  }
}

// ---------------------------------------------------------------------------
extern "C" void kernel_launch(void* const* d_in, const int* in_sizes, int n_in,
                              void* d_out, int out_size, void* d_ws, size_t ws_size,
                              hipStream_t stream) {
  const float*         ego   = (const float*)d_in[0];          // (B,1,D)
  const unsigned char* mask  = (const unsigned char*)d_in[1];  // (B,N) bool
  const float*         keys  = (const float*)d_in[2];          // (H,B,NK,DH)
  const float*         value = (const float*)d_in[3];          // (B,NK,DH)
  const float*         wq    = (const float*)d_in[4];          // (H,D,DH)

  float* q_ws = (float*)d_ws;  // (H,B,DH) fp32 = 512 KB scratch

  qproj_wmma<<<dim3(B_ / 16, H_), 32, 0, stream>>>(ego, wq, q_ws);
  attn_fused<<<B_, 256, 0, stream>>>(q_ws, mask, keys, value, (float*)d_out);
}